// GCNNodeEdge_51951924412646
// MI455X (gfx1250) — compile-verified
//
#include <hip/hip_runtime.h>

#define IN_C    64
#define HID     128
#define HID2    256
#define LID_IN  195
#define OUT_C   24
#define BN_EPS  1e-5f

typedef __attribute__((ext_vector_type(2))) float v2f;
typedef __attribute__((ext_vector_type(8))) float v8f;

// ---------------------------------------------------------------------------
// Fp32 WMMA GEMM: Y[M,NO] = X[M,K] @ W[K,NO]; optional fused bias + ReLU.
// Compile-time K / NO so the K-loop fully unrolls with zero guards
// (only the 195-K lid GEMM keeps a single guarded tail step).
// One wave owns a 16 x (16*NT) output strip: A fragment loaded once per
// K-step feeds NT back-to-back V_WMMA_F32_16X16X4_F32 ops.
// Frag layouts per ISA 7.12.2:
//   A (16x4):  lane m = lane&15; VGPR0/1 = K(k0 + 2*(lane>=16) + {0,1})
//   B (4x16):  lane n = lane&15; same K split
//   C/D:       VGPR v -> row v + 8*(lane>=16), col = lane&15
// ---------------------------------------------------------------------------
template <int K, int NO, int NT, bool BIAS_RELU>
__global__ void gemm_f32_wmma(const float* __restrict__ X,
                              const float* __restrict__ W,
                              const float* __restrict__ bias,
                              float* __restrict__ Y, int M) {
  const int lane  = threadIdx.x & 31;
  const int wave  = threadIdx.x >> 5;
  const int wpb   = blockDim.x >> 5;
  const int mtile = blockIdx.x * wpb + wave;
  if (mtile * 16 >= M) return;                 // wave-uniform: EXEC stays full
  const int n0    = blockIdx.y * (16 * NT);
  const int lcol  = lane & 15;
  const int khalf = (lane >> 4) << 1;          // 0 or 2
  const int mrow  = mtile * 16 + lcol;
  const float* __restrict__ Xr = X + (size_t)mrow * K;

  constexpr bool FULLN = (NO % (16 * NT)) == 0;   // no column guards needed
  constexpr int  KMAIN = (K / 4) * 4;

  v8f acc[NT];
#pragma unroll
  for (int t = 0; t < NT; ++t) acc[t] = (v8f){};

#pragma unroll
  for (int k0 = 0; k0 < KMAIN; k0 += 4) {
    const int ka = k0 + khalf;
    v2f a;
    if constexpr ((K & 3) == 0) {
      a = *reinterpret_cast<const v2f*>(Xr + ka);     // 8B-aligned b64 load
    } else {
      a.x = Xr[ka]; a.y = Xr[ka + 1];
    }
#pragma unroll
    for (int t = 0; t < NT; ++t) {
      const int nc = n0 + t * 16 + lcol;
      const bool okn = FULLN ? true : (nc < NO);
      v2f b;
      b.x = okn ? W[(size_t)(ka)     * NO + nc] : 0.f;
      b.y = okn ? W[(size_t)(ka + 1) * NO + nc] : 0.f;
      acc[t] = __builtin_amdgcn_wmma_f32_16x16x4_f32(false, a, false, b,
                                                     (short)0, acc[t],
                                                     false, false);
    }
  }

  if constexpr (KMAIN < K) {                   // lid tail (K = 195)
    const int ka = KMAIN + khalf;
    v2f a;
    a.x = (ka     < K) ? Xr[ka]     : 0.f;
    a.y = (ka + 1 < K) ? Xr[ka + 1] : 0.f;
#pragma unroll
    for (int t = 0; t < NT; ++t) {
      const int nc = n0 + t * 16 + lcol;
      const bool okn = FULLN ? true : (nc < NO);
      v2f b;
      b.x = (okn && ka     < K) ? W[(size_t)(ka)     * NO + nc] : 0.f;
      b.y = (okn && ka + 1 < K) ? W[(size_t)(ka + 1) * NO + nc] : 0.f;
      acc[t] = __builtin_amdgcn_wmma_f32_16x16x4_f32(false, a, false, b,
                                                     (short)0, acc[t],
                                                     false, false);
    }
  }

  const int rbase = mtile * 16 + ((lane >> 4) << 3);
#pragma unroll
  for (int t = 0; t < NT; ++t) {
    const int nc = n0 + t * 16 + lcol;
    if (FULLN || nc < NO) {
      const float bv = BIAS_RELU ? bias[nc] : 0.f;
#pragma unroll
      for (int v = 0; v < 8; ++v) {
        float val = acc[t][v];
        if (BIAS_RELU) { val += bv; val = val > 0.f ? val : 0.f; }
        Y[(size_t)(rbase + v) * NO + nc] = val;
      }
    }
  }
}

// ---------------------------------------------------------------------------
// Degree / norm precompute (symmetric GCN normalization with self-loops)
// ---------------------------------------------------------------------------
__global__ void deg_accum(const long long* __restrict__ ei,
                          const float* __restrict__ w,
                          float* __restrict__ deg, int E) {
  int e = blockIdx.x * blockDim.x + threadIdx.x;
  if (e < E) atomicAdd(&deg[(int)ei[(size_t)E + e]], w[e]);  // dst row
}

__global__ void deg_finalize(float* __restrict__ deg, int n) {
  int i = blockIdx.x * blockDim.x + threadIdx.x;
  if (i < n) deg[i] = rsqrtf(deg[i] + 1.0f);   // + self-loop weight; deg>=1
}

__global__ void edge_norm(const long long* __restrict__ ei,
                          const float* __restrict__ w,
                          const float* __restrict__ dinv,
                          float* __restrict__ nrm, int E) {
  int e = blockIdx.x * blockDim.x + threadIdx.x;
  if (e < E)
    nrm[e] = dinv[(int)ei[e]] * w[e] * dinv[(int)ei[(size_t)E + e]];
}

// ---------------------------------------------------------------------------
// Aggregation: out[i] = dinv[i]^2 * y[i] + b  (self-loop, contention-free)
// then vectorized atomic scatter of edge messages (L2-resident traffic).
// ---------------------------------------------------------------------------
__global__ void agg_init(const float* __restrict__ y,
                         const float* __restrict__ dinv,
                         const float* __restrict__ bias,
                         float* __restrict__ out, int n, int F) {
  int i = blockIdx.x * blockDim.x + threadIdx.x;
  if (i < n * F) {
    int r = i / F, f = i - r * F;
    float dv = dinv[r];
    out[i] = dv * dv * y[i] + bias[f];
  }
}

__global__ void agg_edges(const float* __restrict__ y,
                          float* __restrict__ out,
                          const long long* __restrict__ ei,
                          const float* __restrict__ nrm, int E, int F4) {
  int wave = blockIdx.x * (blockDim.x >> 5) + (threadIdx.x >> 5);
  int lane = threadIdx.x & 31;
  if (wave >= E) return;
  const int s = (int)ei[wave];
  const int d = (int)ei[(size_t)E + wave];
  const float nm = nrm[wave];
  const float4* __restrict__ yr = (const float4*)(y + (size_t)s * (F4 * 4));
  float* __restrict__ orow = out + (size_t)d * (F4 * 4);
  for (int c = lane; c < F4; c += 32) {
    float4 v = yr[c];                          // global_load_b128 gather
    atomicAdd(&orow[c * 4 + 0], nm * v.x);
    atomicAdd(&orow[c * 4 + 1], nm * v.y);
    atomicAdd(&orow[c * 4 + 2], nm * v.z);
    atomicAdd(&orow[c * 4 + 3], nm * v.w);
  }
}

// ---------------------------------------------------------------------------
// Training-mode BatchNorm: one block per channel, deterministic LDS reduce,
// folded into scale/shift:  x_hat = x*scale + shift.
// ---------------------------------------------------------------------------
__global__ void bn_stats(const float* __restrict__ a,
                         const float* __restrict__ g,
                         const float* __restrict__ be,
                         float* __restrict__ scale,
                         float* __restrict__ shift, int n, int F) {
  const int c = blockIdx.x;
  float s = 0.f, s2 = 0.f;
  for (int r = threadIdx.x; r < n; r += blockDim.x) {
    float v = a[(size_t)r * F + c];
    s += v; s2 += v * v;
  }
  __shared__ float shS[256], shQ[256];
  shS[threadIdx.x] = s; shQ[threadIdx.x] = s2;
  __syncthreads();
  for (int off = blockDim.x >> 1; off > 0; off >>= 1) {
    if (threadIdx.x < off) {
      shS[threadIdx.x] += shS[threadIdx.x + off];
      shQ[threadIdx.x] += shQ[threadIdx.x + off];
    }
    __syncthreads();
  }
  if (threadIdx.x == 0) {
    float invN = 1.f / (float)n;
    float m   = shS[0] * invN;
    float var = shQ[0] * invN - m * m;          // biased variance
    float sc  = g[c] * rsqrtf(var + BN_EPS);
    scale[c] = sc;
    shift[c] = be[c] - m * sc;
  }
}

// Layer-1 fusion: relu(bn(a)) + lid(already relu'd), then zero-mask rows.
__global__ void combine1(const float* __restrict__ a,
                         const float* __restrict__ scale,
                         const float* __restrict__ shift,
                         const float* __restrict__ lid,
                         const float* __restrict__ x_in,
                         float* __restrict__ xo, int n) {
  int i = blockIdx.x * blockDim.x + threadIdx.x;
  if (i < n * HID) {
    int r = i / HID, f = i - r * HID;
    float v = a[i] * scale[f] + shift[f];
    v = v > 0.f ? v : 0.f;
    v += lid[i];
    if (x_in[(size_t)r * IN_C] == 0.f) v = 0.f;
    xo[i] = v;
  }
}

__global__ void bn_relu_apply(const float* __restrict__ a,
                              const float* __restrict__ scale,
                              const float* __restrict__ shift,
                              float* __restrict__ xo, int n, int F) {
  int i = blockIdx.x * blockDim.x + threadIdx.x;
  if (i < n * F) {
    int f = i % F;
    float v = a[i] * scale[f] + shift[f];
    xo[i] = v > 0.f ? v : 0.f;
  }
}

// Final: relu(bn(a4)) -> d_out, plus global sum into d_out[n*OUT_C].
__global__ void final_out(const float* __restrict__ a,
                          const float* __restrict__ scale,
                          const float* __restrict__ shift,
                          float* __restrict__ out,
                          float* __restrict__ sumx, int n) {
  int i = blockIdx.x * blockDim.x + threadIdx.x;
  float v = 0.f;
  if (i < n * OUT_C) {
    int f = i % OUT_C;
    v = a[i] * scale[f] + shift[f];
    v = v > 0.f ? v : 0.f;
    out[i] = v;
  }
  __shared__ float sh[256];
  sh[threadIdx.x] = v;
  __syncthreads();
  for (int off = blockDim.x >> 1; off > 0; off >>= 1) {
    if (threadIdx.x < off) sh[threadIdx.x] += sh[threadIdx.x + off];
    __syncthreads();
  }
  if (threadIdx.x == 0) atomicAdd(sumx, sh[0]);
}

// ---------------------------------------------------------------------------
extern "C" void kernel_launch(void* const* d_in, const int* in_sizes, int n_in,
                              void* d_out, int out_size, void* d_ws, size_t ws_size,
                              hipStream_t stream) {
  const float*     x_in = (const float*)d_in[0];
  const long long* ei   = (const long long*)d_in[1];   // int64 [2,E]
  const float*     ew   = (const float*)d_in[2];
  const float*     lidT = (const float*)d_in[3];
  const float *W1 = (const float*)d_in[4],  *b1  = (const float*)d_in[5];
  const float *Wl = (const float*)d_in[6],  *bl  = (const float*)d_in[7];
  const float *W2 = (const float*)d_in[8],  *b2  = (const float*)d_in[9];
  const float *W3 = (const float*)d_in[10], *b3  = (const float*)d_in[11];
  const float *W4 = (const float*)d_in[12], *b4  = (const float*)d_in[13];
  const float *g1 = (const float*)d_in[14], *be1 = (const float*)d_in[15];
  const float *g2 = (const float*)d_in[16], *be2 = (const float*)d_in[17];
  const float *g3 = (const float*)d_in[18], *be3 = (const float*)d_in[19];
  const float *g4 = (const float*)d_in[20], *be4 = (const float*)d_in[21];

  const int n = in_sizes[0] / IN_C;
  const int E = in_sizes[2];
  float* out  = (float*)d_out;
  float* sumx = out + (size_t)n * OUT_C;

  // ---- workspace carve-out (256B aligned slices) ----
  char* base = (char*)d_ws;
  size_t off = 0;
  auto alloc = [&](size_t bytes) {
    void* p = base + off;
    off = (off + bytes + 255) & ~(size_t)255;
    return p;
  };
  float* dinv  = (float*)alloc((size_t)n * 4);
  float* nrm   = (float*)alloc((size_t)E * 4);
  float* scale = (float*)alloc(HID2 * 4);
  float* shift = (float*)alloc(HID2 * 4);
  float* ybuf  = (float*)alloc((size_t)n * HID2 * 4);
  float* abuf  = (float*)alloc((size_t)n * HID2 * 4);
  float* xbuf  = (float*)alloc((size_t)n * HID2 * 4);
  float* lidb  = (float*)alloc((size_t)n * HID * 4);
  (void)ws_size; (void)n_in; (void)out_size;

  const int TB = 256;
  auto cdiv = [](int a, int b) { return (a + b - 1) / b; };

  // ---- normalization precompute ----
  hipMemsetAsync(dinv, 0, (size_t)n * 4, stream);
  hipMemsetAsync(sumx, 0, 4, stream);
  deg_accum   <<<cdiv(E, TB), TB, 0, stream>>>(ei, ew, dinv, E);
  deg_finalize<<<cdiv(n, TB), TB, 0, stream>>>(dinv, n);
  edge_norm   <<<cdiv(E, TB), TB, 0, stream>>>(ei, ew, dinv, nrm, E);

  const int WPB = 4;                 // waves (16-row M tiles) per block
  const int mtiles = cdiv(n, 16);
  const dim3 blkG(32 * WPB);
  const int gx = cdiv(mtiles, WPB);

  auto agg_bn = [&](const float* b, const float* g, const float* be, int NO) {
    agg_init <<<cdiv(n * NO, TB), TB, 0, stream>>>(ybuf, dinv, b, abuf, n, NO);
    agg_edges<<<cdiv(E, TB / 32), TB, 0, stream>>>(ybuf, abuf, ei, nrm, E, NO / 4);
    bn_stats <<<NO, TB, 0, stream>>>(abuf, g, be, scale, shift, n, NO);
  };

  // ---- layer 1 (K=64 -> 128) + lid branch (K=195 -> 128) + mask ----
  gemm_f32_wmma<IN_C, HID, 4, false>
      <<<dim3(gx, HID / 64), blkG, 0, stream>>>(x_in, W1, nullptr, ybuf, n);
  agg_bn(b1, g1, be1, HID);
  gemm_f32_wmma<LID_IN, HID, 4, true>
      <<<dim3(gx, HID / 64), blkG, 0, stream>>>(lidT, Wl, bl, lidb, n);
  combine1<<<cdiv(n * HID, TB), TB, 0, stream>>>(abuf, scale, shift, lidb,
                                                 x_in, xbuf, n);

  // ---- layer 2 (128 -> 256) ----
  gemm_f32_wmma<HID, HID2, 4, false>
      <<<dim3(gx, HID2 / 64), blkG, 0, stream>>>(xbuf, W2, nullptr, ybuf, n);
  agg_bn(b2, g2, be2, HID2);
  bn_relu_apply<<<cdiv(n * HID2, TB), TB, 0, stream>>>(abuf, scale, shift,
                                                       xbuf, n, HID2);

  // ---- layer 3 (256 -> 128) ----
  gemm_f32_wmma<HID2, HID, 4, false>
      <<<dim3(gx, HID / 64), blkG, 0, stream>>>(xbuf, W3, nullptr, ybuf, n);
  agg_bn(b3, g3, be3, HID);
  bn_relu_apply<<<cdiv(n * HID, TB), TB, 0, stream>>>(abuf, scale, shift,
                                                      xbuf, n, HID);

  // ---- layer 4 (128 -> 24) + output + global sum ----
  gemm_f32_wmma<HID, OUT_C, 2, false>
      <<<dim3(gx, 1), blkG, 0, stream>>>(xbuf, W4, nullptr, ybuf, n);
  agg_bn(b4, g4, be4, OUT_C);
  final_out<<<cdiv(n * OUT_C, TB), TB, 0, stream>>>(abuf, scale, shift,
                                                    out, sumx, n);
}